// FinalPreprocessingLayer_51917564674690
// MI455X (gfx1250) — compile-verified
//
#include <hip/hip_runtime.h>
#include <math.h>

#define BB 32
#define SS 384
#define NLM 543
#define LL 67
#define NF 743

#define OFF_C    0
#define OFF_VEL  134
#define OFF_ACC  268
#define OFF_VM   402
#define OFF_AM   469
#define OFF_CUM  536
#define OFF_ANG  603
#define OFF_DCHG 670
#define OFF_DIST 737

typedef float v2f __attribute__((ext_vector_type(2)));
typedef float v8f __attribute__((ext_vector_type(8)));

__device__ __constant__ int d_sel[LL] = {
  33,133,362,263,61,291,199,419,17,84,314,405,320,307,375,321,308,324,318,
  468,469,470,471,472,473,474,475,476,477,478,479,480,481,482,483,484,485,486,487,488,
  522,523,524,525,526,527,528,529,530,531,532,533,534,535,536,537,538,539,540,541,542,
  11,12,13,14,15,16
};

__device__ __forceinline__ float fin(float v) { return __builtin_isfinite(v) ? v : 0.0f; }

// ---------------- Kernel A: gather + center + scale -> c_ws ----------------
// One wave32 per frame; 8 frames per 256-thread block.
__global__ void kA(const float* __restrict__ x, float* __restrict__ c_ws) {
  int frame = blockIdx.x * (blockDim.x >> 5) + (threadIdx.x >> 5);
  int lane  = threadIdx.x & 31;
  if (frame >= BB * SS) return;

  float vx[3], vy[3];
  float sx = 0.f, sy = 0.f, sq = 0.f;
  size_t fbase = (size_t)frame * (NLM * 3);
#pragma unroll
  for (int i = 0; i < 3; ++i) {
    int l  = lane + 32 * i;
    int li = (l < LL) ? l : 0;
    size_t idx = fbase + (size_t)d_sel[li] * 3;
    float a = x[idx];
    float b = x[idx + 1];
    if (l >= LL) { a = 0.f; b = 0.f; }
    vx[i] = a; vy[i] = b;
    sx += a; sy += b; sq += a * a + b * b;
  }
#pragma unroll
  for (int off = 16; off > 0; off >>= 1) {
    sx += __shfl_xor(sx, off, 32);
    sy += __shfl_xor(sy, off, 32);
    sq += __shfl_xor(sq, off, 32);
  }
  float mx  = sx * (1.0f / LL);
  float my  = sy * (1.0f / LL);
  float ssq = sq - (float)LL * (mx * mx + my * my);     // sum of squares of centered values
  float var = ssq / (float)(LL * 2 - 1);                // ddof=1 over 134 values
  var = var > 0.f ? var : 0.f;
  float sd = sqrtf(var);
  float scale = (sd > 1e-6f) ? (1.0f / sd) : 1.0f;

  float2* cw = (float2*)c_ws;
#pragma unroll
  for (int i = 0; i < 3; ++i) {
    int l = lane + 32 * i;
    if (l < LL) {
      cw[(size_t)frame * LL + l] =
          make_float2((vx[i] - mx) * scale, (vy[i] - my) * scale);
    }
  }
}

// ---------------- Kernel B: temporal features -> out (all but cum) ----------------
__global__ void kB(const float* __restrict__ c_ws, float* __restrict__ vm_ws,
                   float* __restrict__ out) {
  int tid = blockIdx.x * blockDim.x + threadIdx.x;
  const int total = BB * SS * LL;
  if (tid >= total) return;
  int l = tid % LL;
  int f = tid / LL;
  int s = f % SS;

  const float2* cw = (const float2*)c_ws;
  float2 c0  = cw[(size_t)f * LL + l];
  float2 cm1 = make_float2(0.f, 0.f), cm2 = make_float2(0.f, 0.f);
  if (s >= 1) cm1 = cw[(size_t)(f - 1) * LL + l];
  if (s >= 2) cm2 = cw[(size_t)(f - 2) * LL + l];

  float vxv = (s >= 1) ? c0.x - cm1.x : 0.f;
  float vyv = (s >= 1) ? c0.y - cm1.y : 0.f;
  float pxv = (s >= 2) ? cm1.x - cm2.x : 0.f;   // vel[s-1]
  float pyv = (s >= 2) ? cm1.y - cm2.y : 0.f;
  float axv = (s >= 2) ? vxv - pxv : 0.f;
  float ayv = (s >= 2) ? vyv - pyv : 0.f;

  float wx = vxv + 1e-8f, wy = vyv + 1e-8f;
  float vmv = sqrtf(wx * wx + wy * wy);
  float ux = axv + 1e-8f, uy = ayv + 1e-8f;
  float amv = sqrtf(ux * ux + uy * uy);
  float angv = atan2f(vyv + 1e-8f, vxv + 1e-8f);
  float angp = atan2f(pyv + 1e-8f, pxv + 1e-8f);
  float dchg = (s >= 2) ? angv - angp : 0.f;

  size_t ob = (size_t)f * NF;
  out[ob + OFF_C   + 2 * l]     = fin(c0.x);
  out[ob + OFF_C   + 2 * l + 1] = fin(c0.y);
  out[ob + OFF_VEL + 2 * l]     = fin(vxv);
  out[ob + OFF_VEL + 2 * l + 1] = fin(vyv);
  out[ob + OFF_ACC + 2 * l]     = fin(axv);
  out[ob + OFF_ACC + 2 * l + 1] = fin(ayv);
  out[ob + OFF_VM  + l]         = fin(vmv);
  out[ob + OFF_AM  + l]         = fin(amv);
  out[ob + OFF_ANG + l]         = fin(angv);
  out[ob + OFF_DCHG + l]        = fin(dchg);
  vm_ws[(size_t)f * LL + l] = vmv;

  if (l < 6) {
    const int pi[6] = {0, 0, 1, 1, 2, 2};
    const int pj[6] = {1, 2, 2, 3, 3, 4};
    float2 ci = cw[(size_t)f * LL + pi[l]];
    float2 cj = cw[(size_t)f * LL + pj[l]];
    float dx = ci.x - cj.x + 1e-8f, dy = ci.y - cj.y + 1e-8f;
    out[ob + OFF_DIST + l] = fin(sqrtf(dx * dx + dy * dy));
  }
}

// ---------------- Kernel C: exclusive cumsum of vm via WMMA tri-matmul ----------------
// cum(16x16 tile) = StrictLowerTri(16x16) x vm(16x16) + carry, K accumulated in 4 WMMAs.
// One wave per (batch, 16-landmark group). EXEC stays all-ones around WMMA (clamped loads).
__global__ void kC(const float* __restrict__ vm_ws, float* __restrict__ out) {
  int b    = blockIdx.x / 5;
  int lg   = blockIdx.x % 5;
  int lane = threadIdx.x;          // 0..31
  int half = lane >> 4;
  int col  = lane & 15;
  int n    = lg * 16 + col;        // landmark column
  int nl   = (n < LL) ? n : 0;     // clamped address (no EXEC divergence on loads)
  bool valid = (n < LL);
  int row = col;                   // A-matrix row owned by this lane (M)

  float carry = 0.0f;
  for (int t0 = 0; t0 < SS; t0 += 16) {
    v8f d;
#pragma unroll
    for (int r = 0; r < 8; ++r) d[r] = carry;   // C = broadcast carry per column

#pragma unroll
    for (int cc = 0; cc < 4; ++cc) {
      int k0 = 4 * cc + 2 * half;               // lane-half K offset (ISA 16x4 f32 layout)
      v2f a, bb;
      a.x = (float)(k0     < row);              // strict lower triangular
      a.y = (float)(k0 + 1 < row);
      float b0 = vm_ws[((size_t)(b * SS + t0 + k0    )) * LL + nl];
      float b1 = vm_ws[((size_t)(b * SS + t0 + k0 + 1)) * LL + nl];
      bb.x = valid ? b0 : 0.0f;
      bb.y = valid ? b1 : 0.0f;
      d = __builtin_amdgcn_wmma_f32_16x16x4_f32(false, a, false, bb, (short)0, d,
                                                false, false);
    }

#pragma unroll
    for (int r = 0; r < 8; ++r) {
      int m = r + 8 * half;                     // D layout: lanes 0-15 M=0..7, 16-31 M=8..15
      int t = t0 + m;
      if (valid)
        out[((size_t)(b * SS + t)) * NF + OFF_CUM + n] = fin(d[r]);
    }

    // carry' = (carry + sum_{k<15} vm) + vm[15] = D[M=15] + vm[15]
    float dm15o = __shfl_xor(d[7], 16, 32);
    float dm15  = half ? d[7] : dm15o;
    float vm15  = valid ? vm_ws[((size_t)(b * SS + t0 + 15)) * LL + nl] : 0.0f;
    carry = dm15 + vm15;
  }
}

extern "C" void kernel_launch(void* const* d_in, const int* in_sizes, int n_in,
                              void* d_out, int out_size, void* d_ws, size_t ws_size,
                              hipStream_t stream) {
  const float* x = (const float*)d_in[0];
  float* out = (float*)d_out;
  float* c_ws  = (float*)d_ws;                          // BB*SS*LL float2  (~6.6 MB)
  float* vm_ws = c_ws + (size_t)BB * SS * LL * 2;       // BB*SS*LL floats  (~3.3 MB)

  int frames = BB * SS;                                  // 12288
  kA<<<frames / 8, 256, 0, stream>>>(x, c_ws);

  int totalB = BB * SS * LL;                             // 823296
  kB<<<(totalB + 255) / 256, 256, 0, stream>>>(c_ws, vm_ws, out);

  kC<<<BB * 5, 32, 0, stream>>>(vm_ws, out);             // 160 waves, 24 tiles each
}